// SpatialTransformer3D_36240934044016
// MI455X (gfx1250) — compile-verified
//
#include <hip/hip_runtime.h>

// 3D spatial transformer (trilinear resample with multiplicative warp).
// B=2, H=W=D=128, C=4, fp32. Memory-bound gather: image (64 MiB) is fully
// L2-resident on MI455X (192 MB L2), so we stream the warp field and output
// with non-temporal hints and let the 8 corner gathers hit L2.

typedef __attribute__((ext_vector_type(4))) float v4f;

__global__ __launch_bounds__(256) void st3d_trilerp_kernel(
    const float* __restrict__ img,   // (B,128,128,128,4) f32
    const float* __restrict__ tr,    // (B,128,128,128,3) f32
    float*       __restrict__ out,   // (B,128,128,128,4) f32
    int ntot)                        // B*128^3 voxels
{
    int idx = blockIdx.x * 256 + threadIdx.x;
    if (idx >= ntot) return;

    // idx -> (b, i=y-row, j=x-col, k=z)  [all dims = 128]
    int k =  idx        & 127;
    int j = (idx >> 7)  & 127;
    int i = (idx >> 14) & 127;
    int b =  idx >> 21;

    // grid[i,j,k] = (xl[j], yl[i], zl[k]), linspace(-1,1,128)
    const float s = 2.0f / 127.0f;
    float glx = fmaf((float)j, s, -1.0f);
    float gly = fmaf((float)i, s, -1.0f);
    float glz = fmaf((float)k, s, -1.0f);

    // streamed once -> non-temporal loads (don't pollute L2)
    size_t t3 = (size_t)idx * 3;
    float tx = __builtin_nontemporal_load(tr + t3 + 0);
    float ty = __builtin_nontemporal_load(tr + t3 + 1);
    float tz = __builtin_nontemporal_load(tr + t3 + 2);

    // coords = t * grid;  sample pos = 0.5*(c+1)*128 = 64*c + 64
    float x = fmaf(tx * glx, 64.0f, 64.0f);
    float y = fmaf(ty * gly, 64.0f, 64.0f);
    float z = fmaf(tz * glz, 64.0f, 64.0f);

    int x0 = (int)floorf(x);
    int y0 = (int)floorf(y);
    int z0 = (int)floorf(z);
    int x1 = min(max(x0 + 1, 0), 127);
    int y1 = min(max(y0 + 1, 0), 127);
    int z1 = min(max(z0 + 1, 0), 127);
    x0 = min(max(x0, 0), 127);
    y0 = min(max(y0, 0), 127);
    z0 = min(max(z0, 0), 127);

    // weights use the CLIPPED upper corner, exactly as the reference
    float dx = (float)x1 - x;
    float dy = (float)y1 - y;
    float dz = (float)z1 - z;
    float ex = 1.0f - dx, ey = 1.0f - dy, ez = 1.0f - dz;

    // image base for this batch, in float4 (=one voxel's 4 channels) units
    const v4f* __restrict__ ib = (const v4f*)img + ((size_t)b << 21);
    int oy0 = y0 << 14, oy1 = y1 << 14;   // y * 128*128
    int ox0 = x0 << 7,  ox1 = x1 << 7;    // x * 128

    // issue all 8 b128 gathers up front for deep MLP into L2
    v4f c000 = ib[oy0 + ox0 + z0];   // g(y0,x0,z0)
    v4f c100 = ib[oy1 + ox0 + z0];   // g(y1,x0,z0)
    v4f c010 = ib[oy0 + ox1 + z0];   // g(y0,x1,z0)
    v4f c110 = ib[oy1 + ox1 + z0];   // g(y1,x1,z0)
    v4f c001 = ib[oy0 + ox0 + z1];   // g(y0,x0,z1)
    v4f c101 = ib[oy1 + ox0 + z1];   // g(y1,x0,z1)
    v4f c011 = ib[oy0 + ox1 + z1];   // g(y0,x1,z1)
    v4f c111 = ib[oy1 + ox1 + z1];   // g(y1,x1,z1)

    float w000 = dz * dx * dy;
    float w100 = dz * dx * ey;
    float w010 = dz * ex * dy;
    float w110 = dz * ex * ey;
    float w001 = ez * dx * dy;
    float w101 = ez * dx * ey;
    float w011 = ez * ex * dy;
    float w111 = ez * ex * ey;

    v4f r = c000 * w000;
    r = r + c100 * w100;
    r = r + c010 * w010;
    r = r + c110 * w110;
    r = r + c001 * w001;
    r = r + c101 * w101;
    r = r + c011 * w011;
    r = r + c111 * w111;

    // written once, never re-read -> non-temporal b128 store
    __builtin_nontemporal_store(r, (v4f*)out + idx);
}

extern "C" void kernel_launch(void* const* d_in, const int* in_sizes, int n_in,
                              void* d_out, int out_size, void* d_ws, size_t ws_size,
                              hipStream_t stream) {
    const float* img = (const float*)d_in[0];      // (2,128,128,128,4) f32
    const float* tr  = (const float*)d_in[1];      // (2,128,128,128,3) f32
    float* out = (float*)d_out;                    // (2,128,128,128,4) f32

    int ntot = in_sizes[1] / 3;                    // B*OH*OW*OD = 4,194,304
    int blocks = (ntot + 255) / 256;
    st3d_trilerp_kernel<<<blocks, 256, 0, stream>>>(img, tr, out, ntot);
}